// DummyMoEModel_59742995087814
// MI455X (gfx1250) — compile-verified
//
#include <hip/hip_runtime.h>
#include <math.h>

typedef __attribute__((ext_vector_type(2))) float v2f;
typedef __attribute__((ext_vector_type(8))) float v8f;

// ---- f32 WMMA 16x16x4: D = A(16x4) x B(4x16) + C(16x16) ----
static __device__ __forceinline__ v8f wmma4(float a0, float a1,
                                            float b0, float b1, v8f c) {
  v2f a = {a0, a1};
  v2f b = {b0, b1};
  return __builtin_amdgcn_wmma_f32_16x16x4_f32(
      /*neg_a=*/false, a, /*neg_b=*/false, b,
      /*c_mod=*/(short)0, c, /*reuse_a=*/false, /*reuse_b=*/false);
}

// ds_swizzle SWAPX16 (group-of-32, xor_mask=0x10, and_mask=0x1f): swap wave halves
static __device__ __forceinline__ float swap16(float x) {
  return __builtin_bit_cast(
      float, __builtin_amdgcn_ds_swizzle(__builtin_bit_cast(int, x), 0x401f));
}
static __device__ __forceinline__ float xor1(float x)  { return __builtin_bit_cast(float, __builtin_amdgcn_ds_swizzle(__builtin_bit_cast(int, x), 0x041f)); }
static __device__ __forceinline__ float xor2(float x)  { return __builtin_bit_cast(float, __builtin_amdgcn_ds_swizzle(__builtin_bit_cast(int, x), 0x081f)); }
static __device__ __forceinline__ float xor4(float x)  { return __builtin_bit_cast(float, __builtin_amdgcn_ds_swizzle(__builtin_bit_cast(int, x), 0x101f)); }
static __device__ __forceinline__ float xor8(float x)  { return __builtin_bit_cast(float, __builtin_amdgcn_ds_swizzle(__builtin_bit_cast(int, x), 0x201f)); }

static __device__ __forceinline__ float gelu_exact(float v) {
  return 0.5f * v * (1.0f + erff(v * 0.70710678118654752f));
}

// C-layout (vgpr r = rows r / r+8) -> B-layout (chunk c: vgpr 2c = rows 4c/4c+2,
// vgpr 2c+1 = rows 4c+1/4c+3). Swizzles are hoisted OUT of the selects so they
// execute convergently under full EXEC (predicated ds_swizzle would read zeros
// from disabled lanes and also spawn saveexec branches).
static __device__ __forceinline__ void cvtC2B(const float c[8], float b[8],
                                              bool isHi) {
  float s[8];
#pragma unroll
  for (int r = 0; r < 8; ++r) s[r] = swap16(c[r]);
  b[0] = isHi ? s[2] : c[0];
  b[1] = isHi ? s[3] : c[1];
  b[2] = isHi ? s[6] : c[4];
  b[3] = isHi ? s[7] : c[5];
  b[4] = isHi ? c[2] : s[0];
  b[5] = isHi ? c[3] : s[1];
  b[6] = isHi ? c[6] : s[4];
  b[7] = isHi ? c[7] : s[5];
}

__global__ void moe_zero_out(float* out) {
  if (threadIdx.x == 0 && blockIdx.x == 0) out[0] = 0.0f;
}

// One wave handles tiles of 16 tokens; tokens ride the WMMA N dimension so
// all weight matrices live in A-layout VGPRs for the whole kernel.
__global__ __launch_bounds__(256) void moe_wmma_kernel(
    const float* __restrict__ inp,
    const float* __restrict__ W_a, const float* __restrict__ b_a,
    const float* __restrict__ W_b, const float* __restrict__ b_b,
    const float* __restrict__ Wg,  const float* __restrict__ bg,
    const float* __restrict__ We1, const float* __restrict__ be1,
    const float* __restrict__ We2, const float* __restrict__ be2,
    float* __restrict__ out, int nTok) {
  const int lane = threadIdx.x & 31;
  const int m    = lane & 15;     // A-matrix row / B-matrix column (token)
  const int hi   = lane >> 4;     // upper half-wave holds K+2 / rows 8..15
  const bool isHi = (hi != 0);

  // ---------------- weight preload (A-layout, resident in VGPRs) -----------
  float aWa[4];                        // A = W_a^T [16h x 8d]
#pragma unroll
  for (int c = 0; c < 2; ++c) {
    int k0 = 4 * c + hi * 2;
    aWa[2 * c + 0] = W_a[(k0 + 0) * 16 + m];
    aWa[2 * c + 1] = W_a[(k0 + 1) * 16 + m];
  }
  float aWb[8];                        // A = W_b^T [8d(pad16) x 16h]
#pragma unroll
  for (int c = 0; c < 4; ++c) {
    int k0 = 4 * c + hi * 2;
    aWb[2 * c + 0] = (m < 8) ? W_b[(k0 + 0) * 8 + m] : 0.0f;
    aWb[2 * c + 1] = (m < 8) ? W_b[(k0 + 1) * 8 + m] : 0.0f;
  }
  float aWg[4];                        // A = Wg^T [2e(pad16) x 8d]
#pragma unroll
  for (int c = 0; c < 2; ++c) {
    int k0 = 4 * c + hi * 2;
    aWg[2 * c + 0] = (m < 2) ? Wg[(k0 + 0) * 2 + m] : 0.0f;
    aWg[2 * c + 1] = (m < 2) ? Wg[(k0 + 1) * 2 + m] : 0.0f;
  }
  float aE1[2][4];                     // A = We1_e^T [16h x 8d]
#pragma unroll
  for (int e = 0; e < 2; ++e)
#pragma unroll
    for (int c = 0; c < 2; ++c) {
      int k0 = 4 * c + hi * 2;
      aE1[e][2 * c + 0] = We1[e * 128 + (k0 + 0) * 16 + m];
      aE1[e][2 * c + 1] = We1[e * 128 + (k0 + 1) * 16 + m];
    }
  float aE2[2][8];                     // A = We2_e^T [8d(pad16) x 16h]
#pragma unroll
  for (int e = 0; e < 2; ++e)
#pragma unroll
    for (int c = 0; c < 4; ++c) {
      int k0 = 4 * c + hi * 2;
      aE2[e][2 * c + 0] = (m < 8) ? We2[e * 128 + (k0 + 0) * 8 + m] : 0.0f;
      aE2[e][2 * c + 1] = (m < 8) ? We2[e * 128 + (k0 + 1) * 8 + m] : 0.0f;
    }
  // biases in C-layout: vgpr r holds row r (lo lanes) / row r+8 (hi lanes)
  float bav[8], bbv[8], be1v[2][8], be2v[2][8];
#pragma unroll
  for (int r = 0; r < 8; ++r) {
    bav[r] = b_a[hi * 8 + r];
    bbv[r] = isHi ? 0.0f : b_b[r];
#pragma unroll
    for (int e = 0; e < 2; ++e) {
      be1v[e][r] = be1[e * 16 + hi * 8 + r];
      be2v[e][r] = isHi ? 0.0f : be2[e * 8 + r];
    }
  }
  const float bg0 = isHi ? 0.0f : bg[0];
  const float bg1 = isHi ? 0.0f : bg[1];

  const int nTiles = nTok >> 4;
  const int waveId = blockIdx.x * (blockDim.x >> 5) + (threadIdx.x >> 5);
  const int nWaves = gridDim.x * (blockDim.x >> 5);

  float lsum = 0.0f;

  for (int tile = waveId; tile < nTiles; tile += nWaves) {
    // ---- load X^T tile [8d x 16tok] in B-layout: 2x global_load_b64/lane --
    const long tok = (long)tile * 16 + m;
    const float* p = inp + tok * 8 + hi * 2;
    v2f u0 = *(const v2f*)(p);       // (d0, d0+1)
    v2f u1 = *(const v2f*)(p + 4);   // (d0+4, d0+5)

    // ---- GEMM1: H = relu(W_a^T x X^T + b_a) ------------------------------
    v8f h = {bav[0], bav[1], bav[2], bav[3], bav[4], bav[5], bav[6], bav[7]};
    h = wmma4(aWa[0], aWa[1], u0.x, u0.y, h);
    h = wmma4(aWa[2], aWa[3], u1.x, u1.y, h);
    float hc[8];
#pragma unroll
    for (int r = 0; r < 8; ++r) hc[r] = fmaxf(h[r], 0.0f);

    // C-layout -> B-layout (16x16), convergent swizzles
    float hb[8];
    cvtC2B(hc, hb, isHi);

    // ---- GEMM2: x = W_b^T(pad) x H + b_b (rows 0..7 valid, 8..15 == 0) ---
    v8f x2 = {bbv[0], bbv[1], bbv[2], bbv[3], bbv[4], bbv[5], bbv[6], bbv[7]};
    x2 = wmma4(aWb[0], aWb[1], hb[0], hb[1], x2);
    x2 = wmma4(aWb[2], aWb[3], hb[2], hb[3], x2);
    x2 = wmma4(aWb[4], aWb[5], hb[4], hb[5], x2);
    x2 = wmma4(aWb[6], aWb[7], hb[6], hb[7], x2);

    // partial C->B: only K-rows 0..7 (chunks 0,1) needed downstream.
    // Swizzles hoisted out of the selects (convergent, full EXEC).
    float sx0 = swap16(x2[2]);
    float sx1 = swap16(x2[3]);
    float sx2 = swap16(x2[6]);
    float sx3 = swap16(x2[7]);
    float xb0 = isHi ? sx0 : x2[0];
    float xb1 = isHi ? sx1 : x2[1];
    float xb2 = isHi ? sx2 : x2[4];
    float xb3 = isHi ? sx3 : x2[5];

    // ---- gate: logits land in acc[0]/acc[1] of the SAME lane (col=token) -
    v8f g = {bg0, bg1, 0.0f, 0.0f, 0.0f, 0.0f, 0.0f, 0.0f};
    g = wmma4(aWg[0], aWg[1], xb0, xb1, g);
    g = wmma4(aWg[2], aWg[3], xb2, xb3, g);
    float selF = (g[1] > g[0]) ? 1.0f : 0.0f;   // argmax ties -> expert 0
    float selS = swap16(selF);                  // convergent broadcast
    float selB = isHi ? selS : selF;
    const bool sel1 = (selB != 0.0f);

    // ---- both experts' hidden pre-activation -----------------------------
    v8f e0 = {be1v[0][0], be1v[0][1], be1v[0][2], be1v[0][3],
              be1v[0][4], be1v[0][5], be1v[0][6], be1v[0][7]};
    v8f e1 = {be1v[1][0], be1v[1][1], be1v[1][2], be1v[1][3],
              be1v[1][4], be1v[1][5], be1v[1][6], be1v[1][7]};
    e0 = wmma4(aE1[0][0], aE1[0][1], xb0, xb1, e0);
    e0 = wmma4(aE1[0][2], aE1[0][3], xb2, xb3, e0);
    e1 = wmma4(aE1[1][0], aE1[1][1], xb0, xb1, e1);
    e1 = wmma4(aE1[1][2], aE1[1][3], xb2, xb3, e1);

    // top-1 select BEFORE gelu (halves the erf cost), exact gelu
    float gm[8];
#pragma unroll
    for (int r = 0; r < 8; ++r) {
      float v = sel1 ? e1[r] : e0[r];
      gm[r] = gelu_exact(v);
    }

    // C->B of gelu(H_sel), convergent swizzles
    float gb[8];
    cvtC2B(gm, gb, isHi);

    // zero-split per expert (B-layout column == token == lane&15)
    float g0b[8], g1b[8];
#pragma unroll
    for (int r = 0; r < 8; ++r) {
      g1b[r] = sel1 ? gb[r] : 0.0f;
      g0b[r] = sel1 ? 0.0f : gb[r];
    }

    // ---- y = We2_0^T x g0 + We2_1^T x g1 + be2_sel -----------------------
    v8f y;
#pragma unroll
    for (int r = 0; r < 8; ++r) y[r] = sel1 ? be2v[1][r] : be2v[0][r];
    y = wmma4(aE2[0][0], aE2[0][1], g0b[0], g0b[1], y);
    y = wmma4(aE2[0][2], aE2[0][3], g0b[2], g0b[3], y);
    y = wmma4(aE2[0][4], aE2[0][5], g0b[4], g0b[5], y);
    y = wmma4(aE2[0][6], aE2[0][7], g0b[6], g0b[7], y);
    y = wmma4(aE2[1][0], aE2[1][1], g1b[0], g1b[1], y);
    y = wmma4(aE2[1][2], aE2[1][3], g1b[2], g1b[3], y);
    y = wmma4(aE2[1][4], aE2[1][5], g1b[4], g1b[5], y);
    y = wmma4(aE2[1][6], aE2[1][7], g1b[6], g1b[7], y);

    // rows 8..15 (hi-lane entries) are exactly 0 by construction
#pragma unroll
    for (int r = 0; r < 8; ++r) lsum += y[r];
  }

  // ---- wave reduction (swizzle butterflies), block LDS, one atomic -------
  lsum += xor1(lsum);
  lsum += xor2(lsum);
  lsum += xor4(lsum);
  lsum += xor8(lsum);
  lsum += swap16(lsum);
  __shared__ float red[8];
  if (lane == 0) red[threadIdx.x >> 5] = lsum;
  __syncthreads();
  if (threadIdx.x == 0) {
    float t = 0.0f;
#pragma unroll
    for (int i = 0; i < 8; ++i) t += red[i];
    atomicAdd(out, t);
  }
}

extern "C" void kernel_launch(void* const* d_in, const int* in_sizes, int n_in,
                              void* d_out, int out_size, void* d_ws, size_t ws_size,
                              hipStream_t stream) {
  const float* inp = (const float*)d_in[0];
  const float* W_a = (const float*)d_in[1];
  const float* b_a = (const float*)d_in[2];
  const float* W_b = (const float*)d_in[3];
  const float* b_b = (const float*)d_in[4];
  const float* Wg  = (const float*)d_in[5];
  const float* bg  = (const float*)d_in[6];
  const float* We1 = (const float*)d_in[7];
  const float* be1 = (const float*)d_in[8];
  const float* We2 = (const float*)d_in[9];
  const float* be2 = (const float*)d_in[10];
  float* out = (float*)d_out;
  const int nTok = in_sizes[0] / 8;   // N = 4,194,304 (multiple of 16)

  moe_zero_out<<<1, 64, 0, stream>>>(out);
  // 2048 blocks x 8 waves = 16384 waves; 262144 tiles -> 16 tiles/wave
  moe_wmma_kernel<<<2048, 256, 0, stream>>>(inp, W_a, b_a, W_b, b_b, Wg, bg,
                                            We1, be1, We2, be2, out, nTok);
}